// FNO_block_54589034332875
// MI455X (gfx1250) — compile-verified
//
#include <hip/hip_runtime.h>
#include <math.h>

// ---------------------------------------------------------------------------
// Problem constants (B=1, C=32, H=D=W=32)
// ---------------------------------------------------------------------------
#define NC    32
#define NH    32
#define ND    32
#define NW    32
#define NK    17          // rfft length along W
#define NSP   32768       // H*D*W
#define NTOT  1048576     // C*H*D*W
#define CPLX  557056      // C*H*D*17

typedef float v2f __attribute__((ext_vector_type(2)));
typedef float v8f __attribute__((ext_vector_type(8)));

__device__ __forceinline__ float gelu_f(float v) {
    return 0.5f * v * (1.0f + erff(v * 0.70710678118654752440f));
}
__device__ __forceinline__ float sigmoid_f(float v) {
    return 1.0f / (1.0f + expf(-v));
}

// ---------------------------------------------------------------------------
// 1) rfft along W: x (C,H,D,32) real -> (C,H,D,17) complex.  Direct 32-pt DFT.
//    Block = 256 lines staged through LDS (pad 33 to kill bank conflicts).
// ---------------------------------------------------------------------------
__global__ __launch_bounds__(256) void k_rfft_w(const float* __restrict__ x,
                                                float* __restrict__ outr,
                                                float* __restrict__ outi) {
    __shared__ float tile[256 * 33];
    __shared__ float twr[32], twi[32];
    int tid = threadIdx.x;
    if (tid < 32) {
        float th = 6.28318530717958647692f * (float)tid / 32.0f;
        twr[tid] = cosf(th);
        twi[tid] = -sinf(th);          // forward sign
    }
    size_t base = (size_t)blockIdx.x * 256 * 32;
    for (int i = tid; i < 8192; i += 256) {
        int ln = i >> 5, w = i & 31;
        tile[ln * 33 + w] = x[base + i];
    }
    __syncthreads();
    size_t line = (size_t)blockIdx.x * 256 + tid;
    const float* lp = &tile[tid * 33];
    for (int k = 0; k < NK; ++k) {
        float sr = 0.f, si = 0.f;
        #pragma unroll
        for (int w = 0; w < 32; ++w) {
            int j = (w * k) & 31;
            float v = lp[w];
            sr += v * twr[j];
            si += v * twi[j];
        }
        outr[line * NK + k] = sr;
        outi[line * NK + k] = si;
    }
}

// ---------------------------------------------------------------------------
// 2) Generic 32-point complex DFT along one axis of the (C, A, B, 17) array.
//    Block = one (p1,p2) plane of 32 x 17 elements.  sign=-1 fwd, +1 inv.
// ---------------------------------------------------------------------------
__global__ __launch_bounds__(256) void k_dft32(const float* __restrict__ in_r,
                                               const float* __restrict__ in_i,
                                               float* __restrict__ out_r,
                                               float* __restrict__ out_i,
                                               int ps1, int ps2, int line_stride,
                                               float sgn) {
    __shared__ float lr[544], li[544], twr[32], twi[32];
    int tid = threadIdx.x;
    int p1 = blockIdx.x >> 5, p2 = blockIdx.x & 31;
    int base = p1 * ps1 + p2 * ps2;
    if (tid < 32) {
        float th = 6.28318530717958647692f * (float)tid / 32.0f;
        twr[tid] = cosf(th);
        twi[tid] = sgn * sinf(th);
    }
    for (int f = tid; f < 544; f += 256) {
        int l = f / 17, s = f - l * 17;
        lr[f] = in_r[base + l * line_stride + s];
        li[f] = in_i[base + l * line_stride + s];
    }
    __syncthreads();
    for (int f = tid; f < 544; f += 256) {
        int ko = f / 17, s = f - ko * 17;
        float ar = 0.f, ai = 0.f;
        #pragma unroll
        for (int l = 0; l < 32; ++l) {
            int j = (l * ko) & 31;
            float xr = lr[l * 17 + s], xi = li[l * 17 + s];
            float tr = twr[j], ti = twi[j];
            ar += xr * tr - xi * ti;
            ai += xr * ti + xi * tr;
        }
        out_r[base + ko * line_stride + s] = ar;
        out_i[base + ko * line_stride + s] = ai;
    }
}

// ---------------------------------------------------------------------------
// 3) Spectral multiply.  Thread = one mode (corner,mx,my,mz); loops o,i.
//    Weight reads are fully coalesced (mode index is the fastest dimension).
//    128 MB of weights streamed exactly once -> the HBM-bound stage.
// ---------------------------------------------------------------------------
__global__ __launch_bounds__(256) void k_spectral(const float* __restrict__ Ar,
                                                  const float* __restrict__ Ai,
                                                  const float* __restrict__ wr,
                                                  const float* __restrict__ wi,
                                                  float* __restrict__ Br,
                                                  float* __restrict__ Bi) {
    int gid = blockIdx.x * 256 + threadIdx.x;     // 0..16383
    int corner = gid >> 12;
    int rem = gid & 4095;
    int mx = rem >> 8, my = (rem >> 4) & 15, mz = rem & 15;
    int kh = mx + ((corner & 1) << 4);
    int kd = my + (((corner >> 1) & 1) << 4);
    int base_x = (kh * 32 + kd) * NK + mz;

    float xr[32], xi[32];
    #pragma unroll
    for (int i = 0; i < 32; ++i) {
        xr[i] = Ar[i * 17408 + base_x];   // 17408 = CPLX/32 per channel
        xi[i] = Ai[i * 17408 + base_x];
    }
    for (int o = 0; o < 32; ++o) {
        int wbase = (corner * 1024 + o) * 4096 + rem;
        float ar = 0.f, ai = 0.f;
        #pragma unroll
        for (int i = 0; i < 32; ++i) {
            float wrv = wr[wbase + i * 131072];
            float wiv = wi[wbase + i * 131072];
            ar += xr[i] * wrv - xi[i] * wiv;
            ai += xr[i] * wiv + xi[i] * wrv;
        }
        Br[o * 17408 + base_x] = ar;
        Bi[o * 17408 + base_x] = ai;
        if (mz == 0) {                        // kz = 16 plane is zeroed
            Br[o * 17408 + (kh * 32 + kd) * NK + 16] = 0.f;
            Bi[o * 17408 + (kh * 32 + kd) * NK + 16] = 0.f;
        }
    }
}

// ---------------------------------------------------------------------------
// 4) irfft along W with global 1/(H*D*W) scale: (C,H,D,17) -> (C,H,D,32) real.
// ---------------------------------------------------------------------------
__global__ __launch_bounds__(256) void k_irfft_w(const float* __restrict__ inr,
                                                 const float* __restrict__ ini,
                                                 float* __restrict__ out) {
    __shared__ float twr[32], twi[32];
    int tid = threadIdx.x;
    if (tid < 32) {
        float th = 6.28318530717958647692f * (float)tid / 32.0f;
        twr[tid] = cosf(th);
        twi[tid] = sinf(th);               // inverse sign
    }
    __syncthreads();
    size_t line = (size_t)blockIdx.x * 256 + tid;
    float xr[NK], xi[NK];
    #pragma unroll
    for (int k = 0; k < NK; ++k) {
        xr[k] = inr[line * NK + k];
        xi[k] = ini[line * NK + k];
    }
    const float scale = 1.0f / 32768.0f;
    for (int w = 0; w < 32; ++w) {
        float s = xr[0];
        #pragma unroll
        for (int k = 1; k < 16; ++k) {
            int j = (w * k) & 31;
            s += 2.0f * (xr[k] * twr[j] - xi[k] * twi[j]);
        }
        int j16 = (w * 16) & 31;
        s += xr[16] * twr[j16] - xi[16] * twi[j16];
        out[line * 32 + w] = s * scale;
    }
}

// ---------------------------------------------------------------------------
// 5) Global mean/var reduction over 1,048,576 elements (two pass).
// ---------------------------------------------------------------------------
__global__ __launch_bounds__(256) void k_reduce1(const float* __restrict__ in,
                                                 float* __restrict__ psum,
                                                 float* __restrict__ psq) {
    __shared__ float ss[256], sq[256];
    int tid = threadIdx.x;
    size_t i0 = (size_t)blockIdx.x * 1024 + tid;
    float s = 0.f, q = 0.f;
    #pragma unroll
    for (int it = 0; it < 4; ++it) {
        float v = in[i0 + it * 256];
        s += v; q += v * v;
    }
    ss[tid] = s; sq[tid] = q;
    __syncthreads();
    for (int st = 128; st > 0; st >>= 1) {
        if (tid < st) { ss[tid] += ss[tid + st]; sq[tid] += sq[tid + st]; }
        __syncthreads();
    }
    if (tid == 0) { psum[blockIdx.x] = ss[0]; psq[blockIdx.x] = sq[0]; }
}

__global__ __launch_bounds__(1024) void k_reduce2(const float* __restrict__ psum,
                                                  const float* __restrict__ psq,
                                                  float* __restrict__ stats) {
    __shared__ float ss[1024], sq[1024];
    int tid = threadIdx.x;
    ss[tid] = psum[tid]; sq[tid] = psq[tid];
    __syncthreads();
    for (int st = 512; st > 0; st >>= 1) {
        if (tid < st) { ss[tid] += ss[tid + st]; sq[tid] += sq[tid + st]; }
        __syncthreads();
    }
    if (tid == 0) {
        float mu = ss[0] * (1.0f / 1048576.0f);
        float var = sq[0] * (1.0f / 1048576.0f) - mu * mu;
        stats[0] = mu;
        stats[1] = rsqrtf(var + 1e-5f);
    }
}

// ---------------------------------------------------------------------------
// 6) h := gelu( gn(h)*g[c] + b[c] + x )   (in place)
// ---------------------------------------------------------------------------
__global__ __launch_bounds__(256) void k_gnres(float* __restrict__ h,
                                               const float* __restrict__ x,
                                               const float* __restrict__ g,
                                               const float* __restrict__ b,
                                               const float* __restrict__ stats) {
    size_t i = (size_t)blockIdx.x * 256 + threadIdx.x;
    int c = (int)(i >> 15);
    float mu = stats[0], rs = stats[1];
    float v = (h[i] - mu) * rs * g[c] + b[c] + x[i];
    h[i] = gelu_f(v);
}

// ---------------------------------------------------------------------------
// 7) WMMA f32 GEMM (templated): out(MT*16 x 32768) = act(W * in + bias).
//    One wave owns a full 16-column stripe across all MT row-tiles so each
//    B fragment (the dominant HBM stream) is loaded once and fed to MT WMMAs.
//    V_WMMA_F32_16X16X4_F32 layouts per CDNA5 ISA:
//      A 16x4 : v0 = K{0,2}, v1 = K{1,3}  (lanes 16-31 take K+2)
//      C/D    : 8 VGPRs, row = r + 8*(lane>=16), col = lane&15
// ---------------------------------------------------------------------------
template <int CIN, int MT, int ACT>
__global__ __launch_bounds__(256) void k_gemm_t(const float* __restrict__ W,
                                                const float* __restrict__ bias,
                                                const float* __restrict__ in,
                                                float* __restrict__ out) {
    const int NCOL = NSP;
    int wid  = blockIdx.x * 8 + (threadIdx.x >> 5);   // 0..2047 column tiles
    int lane = threadIdx.x & 31;
    int n0 = wid << 4;
    int lh = lane >> 4, lm = lane & 15;
    v8f acc[MT];
    #pragma unroll
    for (int m = 0; m < MT; ++m) acc[m] = (v8f){};
    #pragma unroll
    for (int k0 = 0; k0 < CIN; k0 += 4) {
        int ka = k0 + (lh << 1);
        v2f b;
        b.x = in[(size_t)ka * NCOL + n0 + lm];
        b.y = in[(size_t)(ka + 1) * NCOL + n0 + lm];
        #pragma unroll
        for (int m = 0; m < MT; ++m) {
            v2f a = *(const v2f*)(W + (size_t)(m * 16 + lm) * CIN + ka);
            acc[m] = __builtin_amdgcn_wmma_f32_16x16x4_f32(false, a, false, b,
                                                           (short)0, acc[m],
                                                           false, false);
        }
    }
    int col = n0 + lm;
    #pragma unroll
    for (int m = 0; m < MT; ++m) {
        #pragma unroll
        for (int r = 0; r < 8; ++r) {
            int row = m * 16 + r + (lh << 3);
            float v = acc[m][r] + bias[row];
            if (ACT == 1) v = gelu_f(v);
            out[(size_t)row * NCOL + col] = v;
        }
    }
}

// ---------------------------------------------------------------------------
// 8) Time-conditioned IN params: gamma/beta = tin_*w @ t + tin_*b  (64 rows)
// ---------------------------------------------------------------------------
__global__ __launch_bounds__(64) void k_tinp(const float* __restrict__ gw,
                                             const float* __restrict__ gbias,
                                             const float* __restrict__ bw,
                                             const float* __restrict__ bbias,
                                             const float* __restrict__ t,
                                             float* __restrict__ gb) {
    int r = threadIdx.x;
    if (r < 32) {
        float s = gbias[r];
        for (int c = 0; c < 32; ++c) s += gw[r * 32 + c] * t[c];
        gb[r] = s;
    } else {
        int rr = r - 32;
        float s = bbias[rr];
        for (int c = 0; c < 32; ++c) s += bw[rr * 32 + c] * t[c];
        gb[r] = s;
    }
}

// h2 := gamma[c]*gn(h2) + beta[c]   (in place; stats already computed)
__global__ __launch_bounds__(256) void k_tin(float* __restrict__ h2,
                                             const float* __restrict__ gb,
                                             const float* __restrict__ stats) {
    size_t i = (size_t)blockIdx.x * 256 + threadIdx.x;
    int c = (int)(i >> 15);
    float mu = stats[0], rs = stats[1];
    h2[i] = gb[c] * (h2[i] - mu) * rs + gb[32 + c];
}

// ---------------------------------------------------------------------------
// 9) Spatial attention + memory slots, one-pass online softmax per (c,n).
//    k_sp/v_sp at shift m == (K*h+b) gathered at the shifted voxel (bias at OOB).
// ---------------------------------------------------------------------------
__global__ __launch_bounds__(256) void k_attn(const float* __restrict__ q,
                                              const float* __restrict__ kh,
                                              const float* __restrict__ vh,
                                              const float* __restrict__ kb,
                                              const float* __restrict__ vb,
                                              const float* __restrict__ mk,
                                              const float* __restrict__ mv,
                                              float* __restrict__ outa) {
    int tid = blockIdx.x * 256 + threadIdx.x;      // 0..NTOT-1
    int c = tid >> 15, n = tid & 32767;
    int hh = n >> 10, dd = (n >> 5) & 31, ww = n & 31;
    float qv = q[tid] * 0.17677669529663687f;      // 1/sqrt(32)
    float m = -1e30f, l = 0.f, acc = 0.f;
    #pragma unroll
    for (int s = 0; s < 5; ++s) {                  // memory slots
        float kvv = mk[c * 5 + s], vvv = mv[c * 5 + s];
        float lg = qv * kvv;
        float mn = fmaxf(m, lg);
        float corr = expf(m - mn);
        float p = expf(lg - mn);
        l = l * corr + p; acc = acc * corr + p * vvv; m = mn;
    }
    float kbc = kb[c], vbc = vb[c];
    size_t cbase = (size_t)c << 15;
    for (int di = -1; di <= 1; ++di)
        for (int dj = -1; dj <= 1; ++dj)
            for (int dl = -1; dl <= 1; ++dl) {
                int h2v = hh + di, d2 = dd + dj, w2 = ww + dl;
                bool ok = ((unsigned)h2v < 32u) & ((unsigned)d2 < 32u) & ((unsigned)w2 < 32u);
                float kvv, vvv;
                if (ok) {
                    size_t idx = cbase + (size_t)(((h2v << 5) + d2) << 5) + w2;
                    kvv = kh[idx]; vvv = vh[idx];
                } else { kvv = kbc; vvv = vbc; }
                float lg = qv * kvv;
                float mn = fmaxf(m, lg);
                float corr = expf(m - mn);
                float p = expf(lg - mn);
                l = l * corr + p; acc = acc * corr + p * vvv; m = mn;
            }
    outa[tid] = acc / l;
}

// ---------------------------------------------------------------------------
// 10) Gate GEMM (WMMA, K=64 over concat[h3, attn], MT=2) fused with sigmoid
//     gate, mix, +gelu(x) skip; writes d_out and the pre-skip field for GRU.
//     k0-based operand select is uniform per unrolled chunk -> folds to a
//     constant pointer (no divergent control flow in the hot loop).
// ---------------------------------------------------------------------------
__global__ __launch_bounds__(256) void k_gate(const float* __restrict__ Wg,
                                              const float* __restrict__ bg,
                                              const float* __restrict__ h3,
                                              const float* __restrict__ attn,
                                              const float* __restrict__ x,
                                              float* __restrict__ outf,
                                              float* __restrict__ dout) {
    const int NCOL = NSP;
    int wid  = blockIdx.x * 8 + (threadIdx.x >> 5);   // 0..2047 column tiles
    int lane = threadIdx.x & 31;
    int n0 = wid << 4;
    int lh = lane >> 4, lm = lane & 15;
    v8f acc[2];
    acc[0] = (v8f){}; acc[1] = (v8f){};
    #pragma unroll
    for (int k0 = 0; k0 < 64; k0 += 4) {
        const float* src = (k0 < 32) ? h3 : (attn - (size_t)32 * NCOL);
        int ka = k0 + (lh << 1);
        v2f b;
        b.x = src[(size_t)ka * NCOL + n0 + lm];
        b.y = src[(size_t)(ka + 1) * NCOL + n0 + lm];
        #pragma unroll
        for (int m = 0; m < 2; ++m) {
            v2f a = *(const v2f*)(Wg + (size_t)(m * 16 + lm) * 64 + ka);
            acc[m] = __builtin_amdgcn_wmma_f32_16x16x4_f32(false, a, false, b,
                                                           (short)0, acc[m],
                                                           false, false);
        }
    }
    int col = n0 + lm;
    #pragma unroll
    for (int m = 0; m < 2; ++m) {
        #pragma unroll
        for (int r = 0; r < 8; ++r) {
            int row = m * 16 + r + (lh << 3);
            size_t idx = (size_t)row * NCOL + col;
            float g = sigmoid_f(acc[m][r] + bg[row]);
            float of = g * attn[idx] + (1.0f - g) * h3[idx];
            outf[idx] = of;
            dout[idx] = of + gelu_f(x[idx]);
        }
    }
}

// ---------------------------------------------------------------------------
// 11) Per-channel spatial mean of the gated field (GRU input).
// ---------------------------------------------------------------------------
__global__ __launch_bounds__(256) void k_cmean(const float* __restrict__ outf,
                                               float* __restrict__ m_in) {
    __shared__ float ss[256];
    int c = blockIdx.x, tid = threadIdx.x;
    const float* p = outf + (size_t)c * NSP;
    float s = 0.f;
    for (int i = tid; i < NSP; i += 256) s += p[i];
    ss[tid] = s;
    __syncthreads();
    for (int st = 128; st > 0; st >>= 1) {
        if (tid < st) ss[tid] += ss[tid + st];
        __syncthreads();
    }
    if (tid == 0) m_in[c] = ss[0] * (1.0f / (float)NSP);
}

// ---------------------------------------------------------------------------
// 12) GRUCell on (1,32): two 96x32 matvecs + pointwise gates.
// ---------------------------------------------------------------------------
__global__ __launch_bounds__(128) void k_gru(const float* __restrict__ wih,
                                             const float* __restrict__ whh,
                                             const float* __restrict__ bih,
                                             const float* __restrict__ bhh,
                                             const float* __restrict__ m_in,
                                             const float* __restrict__ prev,
                                             float* __restrict__ outmem) {
    __shared__ float gi[96], gh[96];
    int tid = threadIdx.x;
    if (tid < 96) {
        float a = bih[tid], b = bhh[tid];
        for (int c = 0; c < 32; ++c) {
            a += wih[tid * 32 + c] * m_in[c];
            b += whh[tid * 32 + c] * prev[c];
        }
        gi[tid] = a; gh[tid] = b;
    }
    __syncthreads();
    if (tid < 32) {
        float r  = sigmoid_f(gi[tid] + gh[tid]);
        float z  = sigmoid_f(gi[32 + tid] + gh[32 + tid]);
        float nn = tanhf(gi[64 + tid] + r * gh[64 + tid]);
        outmem[tid] = (1.0f - z) * nn + z * prev[tid];
    }
}

// ---------------------------------------------------------------------------
// Host-side launcher
// ---------------------------------------------------------------------------
extern "C" void kernel_launch(void* const* d_in, const int* in_sizes, int n_in,
                              void* d_out, int out_size, void* d_ws, size_t ws_size,
                              hipStream_t stream) {
    const float* x       = (const float*)d_in[0];
    const float* t       = (const float*)d_in[1];
    const float* prevmem = (const float*)d_in[2];
    const float* wsr     = (const float*)d_in[3];
    const float* wsi     = (const float*)d_in[4];
    const float* n1g     = (const float*)d_in[5];
    const float* n1b     = (const float*)d_in[6];
    const float* mw1     = (const float*)d_in[7];
    const float* mb1     = (const float*)d_in[8];
    const float* mw2     = (const float*)d_in[9];
    const float* mb2     = (const float*)d_in[10];
    const float* tgw     = (const float*)d_in[11];
    const float* tgb     = (const float*)d_in[12];
    const float* tbw     = (const float*)d_in[13];
    const float* tbb     = (const float*)d_in[14];
    const float* qw      = (const float*)d_in[15];
    const float* qb      = (const float*)d_in[16];
    const float* kw      = (const float*)d_in[17];
    const float* kb      = (const float*)d_in[18];
    const float* vw      = (const float*)d_in[19];
    const float* vb      = (const float*)d_in[20];
    const float* memk    = (const float*)d_in[21];
    const float* memv    = (const float*)d_in[22];
    const float* gw      = (const float*)d_in[23];
    const float* gb_     = (const float*)d_in[24];
    const float* gwih    = (const float*)d_in[25];
    const float* gwhh    = (const float*)d_in[26];
    const float* gbih    = (const float*)d_in[27];
    const float* gbhh    = (const float*)d_in[28];
    float* dout = (float*)d_out;

    float* ws = (float*)d_ws;
    float* Ar   = ws;                    // CPLX
    float* Ai   = ws + 557056;
    float* Br   = ws + 1114112;
    float* Bi   = ws + 1671168;
    float* hbuf = ws + 2228224;          // NTOT   (spectral -> h1 -> attn out)
    float* h2   = ws + 3276800;          // NTOT   (mlp out -> h3)
    float* u    = ws + 4325376;          // 16*NSP
    float* qb_f = ws + 4849664;          // NTOT   (q)
    float* khb  = ws + 5898240;          // NTOT   (k field -> outf)
    float* vhb  = ws + 6946816;          // NTOT   (v field)
    float* psum = ws + 7995392;          // 1024
    float* psq  = ws + 7996416;          // 1024
    float* stats= ws + 7997440;          // 4
    float* gbv  = ws + 7997444;          // 64 (gamma|beta)
    float* m_in = ws + 7997508;          // 32
    (void)in_sizes; (void)n_in; (void)out_size; (void)ws_size;

    // ---- spectral conv: rfftn -> mode mixing -> irfftn ----
    k_rfft_w <<<128, 256, 0, stream>>>(x, Ar, Ai);
    k_dft32  <<<1024, 256, 0, stream>>>(Ar, Ai, Br, Bi, 17408, 544, 17, -1.0f);  // fwd D
    k_dft32  <<<1024, 256, 0, stream>>>(Br, Bi, Ar, Ai, 17408, 17, 544, -1.0f);  // fwd H
    k_spectral<<<64, 256, 0, stream>>>(Ar, Ai, wsr, wsi, Br, Bi);
    k_dft32  <<<1024, 256, 0, stream>>>(Br, Bi, Ar, Ai, 17408, 17, 544, 1.0f);   // inv H
    k_dft32  <<<1024, 256, 0, stream>>>(Ar, Ai, Br, Bi, 17408, 544, 17, 1.0f);   // inv D
    k_irfft_w<<<128, 256, 0, stream>>>(Br, Bi, hbuf);

    // ---- GroupNorm + affine + residual + gelu ----
    k_reduce1<<<1024, 256, 0, stream>>>(hbuf, psum, psq);
    k_reduce2<<<1, 1024, 0, stream>>>(psum, psq, stats);
    k_gnres  <<<4096, 256, 0, stream>>>(hbuf, x, n1g, n1b, stats);

    // ---- pointwise MLP (WMMA f32 GEMMs) ----
    k_gemm_t<32, 1, 1><<<256, 256, 0, stream>>>(mw1, mb1, hbuf, u);  // 32->16, gelu
    k_gemm_t<16, 2, 0><<<256, 256, 0, stream>>>(mw2, mb2, u, h2);    // 16->32

    // ---- time-conditioned instance norm ----
    k_reduce1<<<1024, 256, 0, stream>>>(h2, psum, psq);
    k_reduce2<<<1, 1024, 0, stream>>>(psum, psq, stats + 2);
    k_tinp  <<<1, 64, 0, stream>>>(tgw, tgb, tbw, tbb, t, gbv);
    k_tin   <<<4096, 256, 0, stream>>>(h2, gbv, stats + 2);          // h2 = h3

    // ---- q/k/v fields (WMMA, MT=2: B stream read once per field) ----
    k_gemm_t<32, 2, 0><<<256, 256, 0, stream>>>(qw, qb, h2, qb_f);
    k_gemm_t<32, 2, 0><<<256, 256, 0, stream>>>(kw, kb, h2, khb);
    k_gemm_t<32, 2, 0><<<256, 256, 0, stream>>>(vw, vb, h2, vhb);

    // ---- attention (online softmax over 5 memory + 27 spatial slots) ----
    k_attn<<<4096, 256, 0, stream>>>(qb_f, khb, vhb, kb, vb, memk, memv, hbuf);

    // ---- gate GEMM (WMMA, concat K=64) + mix + skip, writes d_out ----
    k_gate<<<256, 256, 0, stream>>>(gw, gb_, h2, hbuf, x, khb, dout);

    // ---- GRU memory update ----
    k_cmean<<<32, 256, 0, stream>>>(khb, m_in);
    k_gru  <<<1, 128, 0, stream>>>(gwih, gwhh, gbih, gbhh, m_in, prevmem,
                                   dout + NTOT);
}